// CompositionAttention_9457517985914
// MI455X (gfx1250) — compile-verified
//
#include <hip/hip_runtime.h>
#include <hip/hip_bf16.h>
#include <math.h>

#define BATCH 256
#define NROI  196
#define RNN   1024
#define ATT   512
#define KC    512          // K-chunk staged in LDS
#define APAD  516          // padded LDS row stride (floats): 4-float pad -> conflict-free b64 frags
                           // row stride 516*4=2064 B (16B-aligned, so b128 async stores land aligned)

typedef float v2f __attribute__((ext_vector_type(2)));
typedef float v8f __attribute__((ext_vector_type(8)));
typedef int   v4i __attribute__((ext_vector_type(4)));

#if __has_builtin(__builtin_amdgcn_global_load_async_to_lds_b128) && \
    __has_builtin(__builtin_amdgcn_s_wait_asynccnt)
#define USE_ASYNC_LDS 1
typedef __attribute__((address_space(1))) v4i g_v4i;   // global (AS1) 16B vector
typedef __attribute__((address_space(3))) v4i l_v4i;   // LDS    (AS3) 16B vector
// AS1 global address == generic address for global memory.
__device__ __forceinline__ g_v4i* to_gbl(const void* p) {
    return (g_v4i*)(unsigned long long)p;
}
// AS3 pointers are 32-bit LDS offsets = low 32 bits of the generic address.
__device__ __forceinline__ l_v4i* to_lds(void* p) {
    return (l_v4i*)(unsigned int)(unsigned long long)p;
}
#else
#define USE_ASYNC_LDS 0
#endif

// ---------------------------------------------------------------------------
// Repack wv_w[ATT][RNN] -> wvP[RNN/2][ATT][2]  (k-pairs contiguous per column)
// B-fragment for lane (col, k-pair) becomes one aligned 8-byte load, coalesced
// across lanes (stride 8B).
// ---------------------------------------------------------------------------
__global__ void pack_wv_kernel(const float* __restrict__ wv_w, float* __restrict__ wvP) {
    int idx = blockIdx.x * blockDim.x + threadIdx.x;      // [0, ATT*RNN)
    if (idx >= ATT * RNN) return;
    int i = idx & 1;                 // element within k-pair
    int c = (idx >> 1) & (ATT - 1);  // column (a index)
    int p = idx >> 10;               // k-pair index
    wvP[idx] = wv_w[c * RNN + 2 * p + i];
}

// ---------------------------------------------------------------------------
// base[b,a] = h[b]·wh_w[a] + ctx[b]·wv_w[a] + wh_b[a] + wv_b[a]
// One wave per output element; coalesced strided reads + shuffle reduction.
// ---------------------------------------------------------------------------
__global__ void base_kernel(const float* __restrict__ h, const float* __restrict__ ctx,
                            const float* __restrict__ wh_w, const float* __restrict__ wv_w,
                            const float* __restrict__ wh_b, const float* __restrict__ wv_b,
                            float* __restrict__ base) {
    int gw   = (blockIdx.x * blockDim.x + threadIdx.x) >> 5;  // global wave id
    int lane = threadIdx.x & 31;
    if (gw >= BATCH * ATT) return;
    int b = gw >> 9;          // /512
    int a = gw & (ATT - 1);
    const float* hr  = h     + (size_t)b * RNN;
    const float* cr  = ctx   + (size_t)b * RNN;
    const float* whr = wh_w  + (size_t)a * RNN;
    const float* wvr = wv_w  + (size_t)a * RNN;
    float s = 0.f;
    for (int r = lane; r < RNN; r += 32)
        s += hr[r] * whr[r] + cr[r] * wvr[r];
    for (int off = 16; off; off >>= 1) s += __shfl_xor(s, off, 32);
    if (lane == 0) base[gw] = s + wh_b[a] + wv_b[a];
}

// ---------------------------------------------------------------------------
// Score kernel: WMMA f32 GEMM (roi @ wv^T) fused with tanh/wa epilogue.
// grid = (BATCH, 13 row-tiles), block = 128 threads (4 waves).
// Each wave owns 8 column tiles (128 cols); together 4 waves cover ATT=512.
// A tile (16 rows x KC) double-buffered in LDS via GLOBAL_LOAD_ASYNC_TO_LDS:
// chunk k+1's HBM->LDS DMA overlaps chunk k's 1024 WMMAs per wave.
// Fragments per V_WMMA_F32_16X16X4_F32 layout: lanes 0-15 hold
// (row=lane, k0/k0+1), lanes 16-31 hold (row, k0+2/k0+3).
// ---------------------------------------------------------------------------
__device__ __forceinline__ void score_compute_chunk(
        const float* __restrict__ Abuf, const float* __restrict__ wvP,
        v8f acc[8], int kc, int wave, int rowA, int colL, int khalf) {
    for (int k0 = 0; k0 < KC; k0 += 4) {
        v2f afrag = *(const v2f*)&Abuf[rowA * APAD + k0 + khalf * 2];
        const int p = ((kc + k0) >> 1) + khalf;           // k-pair row in wvP
        #pragma unroll
        for (int t = 0; t < 8; ++t) {
            int col = (wave * 8 + t) * 16 + colL;
            v2f bfrag = *(const v2f*)&wvP[(size_t)p * (ATT * 2) + col * 2];
            acc[t] = __builtin_amdgcn_wmma_f32_16x16x4_f32(
                false, afrag, false, bfrag, (short)0, acc[t], false, false);
        }
    }
}

__global__ __launch_bounds__(128)
void score_kernel(const float* __restrict__ roi, const float* __restrict__ wvP,
                  const float* __restrict__ base, const float* __restrict__ wa_w,
                  const float* __restrict__ wa_b, float* __restrict__ scores) {
    __shared__ __align__(16) float Atile[2][16 * APAD];   // ~66 KB ping-pong
    __shared__ float baseS[ATT];
    __shared__ float waS[ATT];
    __shared__ float scoreS[16];

    const int b    = blockIdx.x;
    const int rt   = blockIdx.y;
    const int n0   = rt * 16;
    const int tid  = threadIdx.x;
    const int lane = tid & 31;
    const int wave = tid >> 5;

    for (int i = tid; i < ATT; i += 128) {
        baseS[i] = base[(size_t)b * ATT + i];
        waS[i]   = wa_w[i];
    }
    if (tid < 16) scoreS[tid] = 0.f;

    const float* roiB = roi + (size_t)b * NROI * RNN;

    v8f acc[8];
    #pragma unroll
    for (int t = 0; t < 8; ++t) acc[t] = (v8f)(0.f);

    const int khalf = (lane >> 4);   // 0: k0/k0+1, 1: k0+2/k0+3
    const int rowA  = lane & 15;
    const int colL  = lane & 15;

#if USE_ASYNC_LDS
    // --- async double-buffered A staging: HBM -> LDS direct (ASYNCcnt) ---
    // issue(buf, kc): 16 rows x 128 lanes x b128 = 16 async insts per thread
    auto issue = [&](int buf, int kc) {
        #pragma unroll 4
        for (int i = 0; i < 16; ++i) {
            int n = n0 + i; if (n > NROI - 1) n = NROI - 1;   // clamp pad rows
            const void* g = roiB + (size_t)n * RNN + kc + tid * 4;
            void*       l = &Atile[buf][i * APAD + tid * 4];
            __builtin_amdgcn_global_load_async_to_lds_b128(to_gbl(g), to_lds(l), 0, 0);
        }
    };
    issue(0, 0);
    __builtin_amdgcn_s_wait_asynccnt(0);
    __syncthreads();                       // buf0 visible to all waves
    issue(1, KC);                          // prefetch chunk 1 during chunk-0 math
    score_compute_chunk(Atile[0], wvP, acc, 0, wave, rowA, colL, khalf);
    __builtin_amdgcn_s_wait_asynccnt(0);
    __syncthreads();                       // buf1 visible
    score_compute_chunk(Atile[1], wvP, acc, KC, wave, rowA, colL, khalf);
#else
    // --- fallback: manual VGPR round-trip staging ---
    for (int kc = 0; kc < RNN; kc += KC) {
        __syncthreads();
        for (int i = 0; i < 16; ++i) {
            int n = n0 + i; if (n > NROI - 1) n = NROI - 1;
            float4 v = ((const float4*)(roiB + (size_t)n * RNN + kc))[tid];
            float* dst = &Atile[0][i * APAD + tid * 4];
            dst[0] = v.x; dst[1] = v.y; dst[2] = v.z; dst[3] = v.w;
        }
        __syncthreads();
        score_compute_chunk(Atile[0], wvP, acc, kc, wave, rowA, colL, khalf);
    }
#endif

    // Epilogue: part[j] = sum over this wave's 128 cols of wa[col]*tanh(base-acc)
    float part[8];
    #pragma unroll
    for (int j = 0; j < 8; ++j) part[j] = 0.f;
    #pragma unroll
    for (int t = 0; t < 8; ++t) {
        int col   = (wave * 8 + t) * 16 + colL;
        float wa  = waS[col];
        float bv  = baseS[col];
        #pragma unroll
        for (int j = 0; j < 8; ++j)
            part[j] += wa * tanhf(bv - acc[t][j]);
    }
    // Reduce over the 16 lanes holding each row (halves reduce independently).
    #pragma unroll
    for (int j = 0; j < 8; ++j) {
        float v = part[j];
        v += __shfl_xor(v, 1, 32);
        v += __shfl_xor(v, 2, 32);
        v += __shfl_xor(v, 4, 32);
        v += __shfl_xor(v, 8, 32);
        part[j] = v;
    }
    if (lane == 0 || lane == 16) {
        int rbase = (lane >> 4) * 8;     // lane0 -> rows 0-7, lane16 -> rows 8-15
        #pragma unroll
        for (int j = 0; j < 8; ++j) atomicAdd(&scoreS[rbase + j], part[j]);
    }
    __syncthreads();
    if (tid < 16) {
        int n = n0 + tid;
        if (n < NROI) scores[(size_t)b * NROI + n] = scoreS[tid] + wa_b[0];
    }
}

// ---------------------------------------------------------------------------
// Softmax over N per batch. block = 256 threads, one block per batch.
// ---------------------------------------------------------------------------
__global__ void softmax_kernel(const float* __restrict__ scores, float* __restrict__ weight) {
    const int b = blockIdx.x, tid = threadIdx.x;
    __shared__ float red[8];
    float s = (tid < NROI) ? scores[(size_t)b * NROI + tid] : -3.4e38f;
    float m = s;
    for (int off = 16; off; off >>= 1) m = fmaxf(m, __shfl_xor(m, off, 32));
    if ((tid & 31) == 0) red[tid >> 5] = m;
    __syncthreads();
    float bm = red[0];
    #pragma unroll
    for (int i = 1; i < 8; ++i) bm = fmaxf(bm, red[i]);
    float e = (tid < NROI) ? expf(s - bm) : 0.f;
    float t = e;
    for (int off = 16; off; off >>= 1) t += __shfl_xor(t, off, 32);
    __syncthreads();
    if ((tid & 31) == 0) red[tid >> 5] = t;
    __syncthreads();
    float sum = 0.f;
    #pragma unroll
    for (int i = 0; i < 8; ++i) sum += red[i];
    if (tid < NROI) weight[(size_t)b * NROI + tid] = e / sum;
}

// ---------------------------------------------------------------------------
// comp[b,r] = (sum_w)*ctx[b,r] - sum_n w[b,n]*roi[b,n,r]; streams roi at HBM rate.
// block = 256 threads (one float4 per thread covers RNN=1024), one block per batch.
// ---------------------------------------------------------------------------
__global__ void compose_kernel(const float* __restrict__ roi, const float* __restrict__ ctx,
                               const float* __restrict__ weight, float* __restrict__ comp) {
    const int b = blockIdx.x, tid = threadIdx.x;
    __shared__ float wS[NROI];
    if (tid < NROI) wS[tid] = weight[(size_t)b * NROI + tid];
    __syncthreads();
    const float4* roiB = (const float4*)(roi + (size_t)b * NROI * RNN);
    float4 acc = make_float4(0.f, 0.f, 0.f, 0.f);
    float sw = 0.f;
    for (int n = 0; n < NROI; ++n) {
        float w = wS[n];
        sw += w;
        float4 v = roiB[n * (RNN / 4) + tid];
        acc.x += w * v.x; acc.y += w * v.y; acc.z += w * v.z; acc.w += w * v.w;
    }
    float4 cv = ((const float4*)(ctx + (size_t)b * RNN))[tid];
    float4 out;
    out.x = cv.x * sw - acc.x;
    out.y = cv.y * sw - acc.y;
    out.z = cv.z * sw - acc.z;
    out.w = cv.w * sw - acc.w;
    ((float4*)(comp + (size_t)b * RNN))[tid] = out;
}

// ---------------------------------------------------------------------------
extern "C" void kernel_launch(void* const* d_in, const int* in_sizes, int n_in,
                              void* d_out, int out_size, void* d_ws, size_t ws_size,
                              hipStream_t stream) {
    const float* h    = (const float*)d_in[0];
    const float* roi  = (const float*)d_in[1];
    const float* ctx  = (const float*)d_in[2];
    const float* wh_w = (const float*)d_in[3];
    const float* wh_b = (const float*)d_in[4];
    const float* wv_w = (const float*)d_in[5];
    const float* wv_b = (const float*)d_in[6];
    const float* wa_w = (const float*)d_in[7];
    const float* wa_b = (const float*)d_in[8];

    float* out_comp   = (float*)d_out;                       // [B, RNN]
    float* out_weight = (float*)d_out + BATCH * RNN;         // [B, N]

    // workspace layout (floats): wvP [RNN/2*ATT*2] | base [B*ATT] | scores [B*N]
    float* wvP    = (float*)d_ws;
    float* base   = wvP + (size_t)ATT * RNN;
    float* scores = base + (size_t)BATCH * ATT;

    pack_wv_kernel<<<(ATT * RNN + 255) / 256, 256, 0, stream>>>(wv_w, wvP);
    base_kernel<<<(BATCH * ATT * 32 + 255) / 256, 256, 0, stream>>>(
        h, ctx, wh_w, wv_w, wh_b, wv_b, base);
    dim3 sgrid(BATCH, (NROI + 15) / 16);
    score_kernel<<<sgrid, 128, 0, stream>>>(roi, wvP, base, wa_w, wa_b, scores);
    softmax_kernel<<<BATCH, 256, 0, stream>>>(scores, out_weight);
    compose_kernel<<<BATCH, 256, 0, stream>>>(roi, ctx, out_weight, out_comp);
}